// GCNnet_42142219108853
// MI455X (gfx1250) — compile-verified
//
#include <hip/hip_runtime.h>
#include <hip/hip_bf16.h>

// ---------------------------------------------------------------------------
// GCN forward for MI455X (gfx1250). fp32 everywhere; GEMMs use
// V_WMMA_F32_16X16X4_F32 (exact fp32 via the matrix pipe). Scatter phase is
// the bottleneck (L2-resident gather + f32 atomics); it is wave-per-edge with
// 128-bit gathers.
// ---------------------------------------------------------------------------

#define N_NODES   100000
#define N_EDGES   1600000
#define NUM_GRAPHS 512
#define BN_EPS    1e-5f

typedef __attribute__((ext_vector_type(2))) float v2f;
typedef __attribute__((ext_vector_type(8))) float v8f;

// ---------------- GEMM: C[nrows x F] = A[nrows x K] @ W[K x F] -------------
// One wave computes one 16x16 tile of C; block = (F/16) waves covering all
// columns of a 16-row stripe. grid.x = nrows/16 (exact). EXEC all ones.
template <int K, int F>
__global__ void gemm_f32_wmma(const float* __restrict__ A,
                              const float* __restrict__ W,
                              float* __restrict__ C) {
  const int lane = threadIdx.x & 31;
  const int wave = threadIdx.x >> 5;
  const int row0 = blockIdx.x << 4;
  const int col0 = wave << 4;
  const int m    = lane & 15;           // M (and N) index within tile
  const int koff = (lane >> 4) << 1;    // lanes 16-31 carry K+2,K+3

  const float* arow = A + (size_t)(row0 + m) * K;
  v8f acc = {};
#pragma unroll
  for (int k0 = 0; k0 < K; k0 += 4) {
    // A 16x4 fp32 fragment: VGPR0=K+koff, VGPR1=K+koff+1  (8B-aligned pair)
    v2f a = *(const v2f*)(arow + k0 + koff);
    // B 4x16 fp32 fragment: N striped across lanes, K pair across VGPRs
    v2f b;
    b.x = W[(size_t)(k0 + koff) * F + col0 + m];
    b.y = W[(size_t)(k0 + koff + 1) * F + col0 + m];
    acc = __builtin_amdgcn_wmma_f32_16x16x4_f32(
        /*neg_a=*/false, a, /*neg_b=*/false, b,
        /*c_mod=*/(short)0, acc, /*reuse_a=*/false, /*reuse_b=*/false);
  }
  // D layout: VGPR r -> rows r (lanes 0-15) and r+8 (lanes 16-31)
  const int ms = (lane >> 4) << 3;
#pragma unroll
  for (int r = 0; r < 8; ++r)
    C[(size_t)(row0 + ms + r) * F + col0 + m] = acc[r];
}

// ---------------- elementwise helpers --------------------------------------
__global__ void zero_f4(float4* __restrict__ p, int n4) {
  int i = blockIdx.x * blockDim.x + threadIdx.x;
  if (i < n4) p[i] = make_float4(0.f, 0.f, 0.f, 0.f);
}

__global__ void deg_init(float* __restrict__ deg, int n) {
  int i = blockIdx.x * blockDim.x + threadIdx.x;
  if (i < n) deg[i] = 1.0f;  // self-loop weight
}

__global__ void deg_accum(const int* __restrict__ col,
                          const float* __restrict__ ew,
                          float* __restrict__ deg, int e) {
  int i = blockIdx.x * blockDim.x + threadIdx.x;
  if (i < e) atomicAdd(&deg[col[i]], ew[i]);
}

__global__ void deg_to_dinv(float* __restrict__ deg, int n) {
  int i = blockIdx.x * blockDim.x + threadIdx.x;
  if (i < n) {
    float d = deg[i];
    deg[i] = d > 0.f ? rsqrtf(d) : 0.f;
  }
}

// ---------------- edge scatter: out[c] += h[r] * dinv[r]*w*dinv[c] ---------
// One wave per edge; each lane moves FPL consecutive features (vector load +
// FPL global f32 atomics). F = 32*FPL.
template <int FPL>
__global__ void scatter_edges(const int* __restrict__ row,
                              const int* __restrict__ col,
                              const float* __restrict__ ew,
                              const float* __restrict__ dinv,
                              const float* __restrict__ h,
                              float* __restrict__ out, int E) {
  int t = blockIdx.x * blockDim.x + threadIdx.x;
  int e = t >> 5;
  int lane = t & 31;
  if (e >= E) return;
  const int r = row[e], c = col[e];
  const float norm = dinv[r] * ew[e] * dinv[c];
  constexpr int F = FPL * 32;
  const float* hr = h + (size_t)r * F + lane * FPL;
  float* oc = out + (size_t)c * F + lane * FPL;
  if constexpr (FPL == 4) {
    float4 v = *(const float4*)hr;
    atomicAdd(oc + 0, v.x * norm);
    atomicAdd(oc + 1, v.y * norm);
    atomicAdd(oc + 2, v.z * norm);
    atomicAdd(oc + 3, v.w * norm);
  } else {
    float2 v = *(const float2*)hr;
    atomicAdd(oc + 0, v.x * norm);
    atomicAdd(oc + 1, v.y * norm);
  }
}

// out[n][f] += t[n][f]*dinv[n]^2 + b[f]   (self-loop message + bias)
template <int F>
__global__ void finalize_layer(const float* __restrict__ t,
                               const float* __restrict__ dinv,
                               const float* __restrict__ b,
                               float* __restrict__ out, int n) {
  int i = blockIdx.x * blockDim.x + threadIdx.x;
  if (i >= n * F) return;
  int node = i / F, f = i - node * F;
  float di = dinv[node];
  out[i] += t[i] * (di * di) + b[f];
}

// ---------------- BatchNorm (training mode, biased var) --------------------
__global__ void bn_stats(const float* __restrict__ h,
                         float* __restrict__ stats /*[256]=sum,sumsq*/, int n) {
  __shared__ float ssum[128], ssq[128];
  const int f = threadIdx.x & 127;
  const int half = threadIdx.x >> 7;
  if (threadIdx.x < 128) { ssum[f] = 0.f; ssq[f] = 0.f; }
  __syncthreads();
  float s = 0.f, q = 0.f;
  for (int r = blockIdx.x * 2 + half; r < n; r += gridDim.x * 2) {
    float v = h[(size_t)r * 128 + f];
    s += v; q += v * v;
  }
  atomicAdd(&ssum[f], s);
  atomicAdd(&ssq[f], q);
  __syncthreads();
  if (threadIdx.x < 128) {
    atomicAdd(&stats[f], ssum[f]);
    atomicAdd(&stats[128 + f], ssq[f]);
  }
}

__global__ void bn_apply(float* __restrict__ h,
                         const float* __restrict__ stats,
                         const float* __restrict__ gamma,
                         const float* __restrict__ beta, int n) {
  int i = blockIdx.x * blockDim.x + threadIdx.x;
  if (i >= n * 128) return;
  int f = i & 127;
  const float invN = 1.0f / (float)n;
  float mean = stats[f] * invN;
  float var = stats[128 + f] * invN - mean * mean;
  float sc = gamma[f] * rsqrtf(var + BN_EPS);
  h[i] = (h[i] - mean) * sc + beta[f];
}

// ---------------- relu + per-graph max pool --------------------------------
__global__ void pool_max(const float* __restrict__ h,
                         const int* __restrict__ batch,
                         unsigned* __restrict__ pooled, int n) {
  int i = blockIdx.x * blockDim.x + threadIdx.x;
  if (i >= n * 64) return;
  int node = i >> 6, f = i & 63;
  float v = h[i];
  v = v > 0.f ? v : 0.f;                       // relu; also handles -inf guard
  atomicMax(&pooled[(size_t)batch[node] * 64 + f], __float_as_uint(v));
}

// ---------------- FC head: relu(pooled@W1+b1)@W2+b2 ------------------------
__global__ void fc_head(const float* __restrict__ pooled,
                        const float* __restrict__ fcW1,
                        const float* __restrict__ fcb1,
                        const float* __restrict__ fcW2,
                        const float* __restrict__ fcb2,
                        float* __restrict__ out) {
  int g = blockIdx.x * blockDim.x + threadIdx.x;
  if (g >= NUM_GRAPHS) return;
  const float* p = pooled + (size_t)g * 64;
  float acc = fcb2[0];
#pragma unroll 4
  for (int j = 0; j < 32; ++j) {
    float s = fcb1[j];
    for (int k = 0; k < 64; ++k) s += p[k] * fcW1[k * 32 + j];
    s = s > 0.f ? s : 0.f;
    acc += s * fcW2[j];
  }
  out[g] = acc;
}

// ---------------------------------------------------------------------------
extern "C" void kernel_launch(void* const* d_in, const int* in_sizes, int n_in,
                              void* d_out, int out_size, void* d_ws, size_t ws_size,
                              hipStream_t stream) {
  const float* x    = (const float*)d_in[0];
  const int*   eidx = (const int*)d_in[1];          // [2 x E]
  const float* ew   = (const float*)d_in[2];
  const int*   bat  = (const int*)d_in[3];
  const float* W1 = (const float*)d_in[4],  *b1 = (const float*)d_in[5];
  const float* g1 = (const float*)d_in[6],  *be1 = (const float*)d_in[7];
  const float* W2 = (const float*)d_in[8],  *b2 = (const float*)d_in[9];
  const float* g2 = (const float*)d_in[10], *be2 = (const float*)d_in[11];
  const float* W3 = (const float*)d_in[12], *b3 = (const float*)d_in[13];
  const float* fcW1 = (const float*)d_in[14], *fcb1 = (const float*)d_in[15];
  const float* fcW2 = (const float*)d_in[16], *fcb2 = (const float*)d_in[17];
  float* out = (float*)d_out;

  const int N = N_NODES, E = N_EDGES;
  const int* row = eidx;
  const int* col = eidx + E;

  // workspace carve-up (256B aligned)
  char* ws = (char*)d_ws;
  size_t off = 0;
  auto alloc = [&](size_t bytes) {
    size_t o = off;
    off += (bytes + 255) & ~(size_t)255;
    return o;
  };
  float*    dinv   = (float*)(ws + alloc((size_t)N * 4));
  float*    bufT   = (float*)(ws + alloc((size_t)N * 128 * 4)); // GEMM out
  float*    bufO   = (float*)(ws + alloc((size_t)N * 128 * 4)); // aggregated
  float*    stats  = (float*)(ws + alloc(256 * 4));
  unsigned* pooled = (unsigned*)(ws + alloc((size_t)NUM_GRAPHS * 64 * 4));

  const int TB = 256;
  const int gN   = (N + TB - 1) / TB;
  const int gE   = (E + TB - 1) / TB;
  const int gE32 = (E * 32 + TB - 1) / TB;   // wave per edge

  // ---- degree -> dinv (shared by all layers) ----
  deg_init<<<gN, TB, 0, stream>>>(dinv, N);
  deg_accum<<<gE, TB, 0, stream>>>(col, ew, dinv, E);
  deg_to_dinv<<<gN, TB, 0, stream>>>(dinv, N);

  // ---- layer 1: x[Nx32] @ W1 -> 128, aggregate, BN ----
  gemm_f32_wmma<32, 128><<<N / 16, 256, 0, stream>>>(x, W1, bufT);
  zero_f4<<<(N * 128 / 4 + TB - 1) / TB, TB, 0, stream>>>((float4*)bufO, N * 128 / 4);
  scatter_edges<4><<<gE32, TB, 0, stream>>>(row, col, ew, dinv, bufT, bufO, E);
  finalize_layer<128><<<(N * 128 + TB - 1) / TB, TB, 0, stream>>>(bufT, dinv, b1, bufO, N);
  zero_f4<<<1, 64, 0, stream>>>((float4*)stats, 64);
  bn_stats<<<256, 256, 0, stream>>>(bufO, stats, N);
  bn_apply<<<(N * 128 + TB - 1) / TB, TB, 0, stream>>>(bufO, stats, g1, be1, N);

  // ---- layer 2: h[Nx128] @ W2 -> 128, aggregate, BN ----
  gemm_f32_wmma<128, 128><<<N / 16, 256, 0, stream>>>(bufO, W2, bufT);
  zero_f4<<<(N * 128 / 4 + TB - 1) / TB, TB, 0, stream>>>((float4*)bufO, N * 128 / 4);
  scatter_edges<4><<<gE32, TB, 0, stream>>>(row, col, ew, dinv, bufT, bufO, E);
  finalize_layer<128><<<(N * 128 + TB - 1) / TB, TB, 0, stream>>>(bufT, dinv, b2, bufO, N);
  zero_f4<<<1, 64, 0, stream>>>((float4*)stats, 64);
  bn_stats<<<256, 256, 0, stream>>>(bufO, stats, N);
  bn_apply<<<(N * 128 + TB - 1) / TB, TB, 0, stream>>>(bufO, stats, g2, be2, N);

  // ---- layer 3: h[Nx128] @ W3 -> 64, aggregate ----
  gemm_f32_wmma<128, 64><<<N / 16, 128, 0, stream>>>(bufO, W3, bufT);
  zero_f4<<<(N * 64 / 4 + TB - 1) / TB, TB, 0, stream>>>((float4*)bufO, N * 64 / 4);
  scatter_edges<2><<<gE32, TB, 0, stream>>>(row, col, ew, dinv, bufT, bufO, E);
  finalize_layer<64><<<(N * 64 + TB - 1) / TB, TB, 0, stream>>>(bufT, dinv, b3, bufO, N);

  // ---- relu + global max pool + FC head ----
  zero_f4<<<(NUM_GRAPHS * 64 / 4 + TB - 1) / TB, TB, 0, stream>>>((float4*)pooled,
                                                                  NUM_GRAPHS * 64 / 4);
  pool_max<<<(N * 64 + TB - 1) / TB, TB, 0, stream>>>(bufO, bat, pooled, N);
  fc_head<<<(NUM_GRAPHS + TB - 1) / TB, TB, 0, stream>>>((const float*)pooled,
                                                         fcW1, fcb1, fcW2, fcb2, out);
}